// NonDominatedSelectionLayer_70927089926370
// MI455X (gfx1250) — compile-verified
//
#include <hip/hip_runtime.h>

typedef float v2f __attribute__((ext_vector_type(2)));
typedef float v8f __attribute__((ext_vector_type(8)));

#define B_     8
#define T_     2048
#define N_     4096
#define TSPLIT 8
#define TCHUNK (T_ / TSPLIT)   // 256 time steps per chunk

// ---------------------------------------------------------------------------
// Phase 1: partial sum / sum-of-squares over a T-chunk, 32 assets per wave.
// Reduction is computed on the matrix pipe: D = ones(16x4) * Xtile(4x16) + C.
// With A = all-ones, every row of D is the per-column (per-asset) sum, and the
// result is invariant to B's row->VGPR permutation, so no shuffles are needed.
// ---------------------------------------------------------------------------
__global__ __launch_bounds__(256) void phase1_partials(
    const float* __restrict__ x,
    float* __restrict__ sumPart,
    float* __restrict__ sqPart) {
  const int lane = threadIdx.x & 31;
  const int wave = threadIdx.x >> 5;
  const int nb   = blockIdx.x & 15;          // n-block within batch (0..15)
  const int b    = blockIdx.x >> 4;          // batch
  const int tc   = blockIdx.y;               // t-chunk
  const int n0   = nb * 256 + wave * 32;     // first asset of this wave
  const int col  = lane & 15;                // column within 16-wide tile
  const int rsel = (lane >> 4) << 1;         // rows 0/1 (lanes 0-15) or 2/3

  const size_t rs = (size_t)N_;              // row stride (elements)
  const float* p0 =
      x + ((size_t)b * T_ + (size_t)tc * TCHUNK + rsel) * rs + n0 + col;
  const float* p1 = p0 + 16;                 // second 16-asset tile

  v2f ones; ones.x = 1.0f; ones.y = 1.0f;    // A = all-ones 16x4
  v8f s0 = {}; v8f s1 = {};                  // sum accumulators (2 tiles)
  v8f q0 = {}; v8f q1 = {};                  // sum-of-squares accumulators

#pragma unroll 4
  for (int step = 0; step < TCHUNK / 4; ++step) {
    // Each lane supplies 2 time-samples of its asset column; the two lane
    // halves cover the other 2 rows of the 4x16 B tile.
    float a0 = __builtin_nontemporal_load(p0);
    float a1 = __builtin_nontemporal_load(p0 + rs);
    float c0 = __builtin_nontemporal_load(p1);
    float c1 = __builtin_nontemporal_load(p1 + rs);
    __builtin_prefetch(p0 + 16 * rs, 0, 0);  // 4 steps ahead (global_prefetch)

    v2f b0; b0.x = a0; b0.y = a1;
    v2f b1; b1.x = c0; b1.y = c1;
    v2f b0q = b0 * b0;
    v2f b1q = b1 * b1;

    s0 = __builtin_amdgcn_wmma_f32_16x16x4_f32(false, ones, false, b0,
                                               (short)0, s0, false, false);
    s1 = __builtin_amdgcn_wmma_f32_16x16x4_f32(false, ones, false, b1,
                                               (short)0, s1, false, false);
    q0 = __builtin_amdgcn_wmma_f32_16x16x4_f32(false, ones, false, b0q,
                                               (short)0, q0, false, false);
    q1 = __builtin_amdgcn_wmma_f32_16x16x4_f32(false, ones, false, b1q,
                                               (short)0, q1, false, false);

    p0 += 4 * rs;
    p1 += 4 * rs;
  }

  // Lanes 0-15: tile0, column = lane. Lanes 16-31: tile1, column = lane-16
  // (row 8 of D equals the same column sums since all rows are identical).
  float s = (lane < 16) ? s0[0] : s1[0];
  float q = (lane < 16) ? q0[0] : q1[0];
  size_t outIdx = ((size_t)tc * B_ + b) * N_ + n0 + lane;
  sumPart[outIdx] = s;
  sqPart[outIdx]  = q;
}

// ---------------------------------------------------------------------------
// Phase 2: fixed-order combine of T-chunk partials -> mean / std (ddof=0).
// ---------------------------------------------------------------------------
__global__ __launch_bounds__(256) void phase2_combine(
    const float* __restrict__ sumPart, const float* __restrict__ sqPart,
    float* __restrict__ meanArr, float* __restrict__ riskArr) {
  int idx = blockIdx.x * 256 + threadIdx.x;  // 0 .. B*N-1
  float s = 0.f, q = 0.f;
#pragma unroll
  for (int tcn = 0; tcn < TSPLIT; ++tcn) {
    s += sumPart[(size_t)tcn * B_ * N_ + idx];
    q += sqPart[(size_t)tcn * B_ * N_ + idx];
  }
  float m   = s * (1.0f / T_);
  float var = q * (1.0f / T_) - m * m;
  var = var < 0.f ? 0.f : var;
  meanArr[idx] = m;
  riskArr[idx] = sqrtf(var);
}

// ---------------------------------------------------------------------------
// Phase 3: dominance check. One batch's mean/risk (32 KiB) staged in LDS;
// each thread scans all j with b128 LDS broadcasts + wave-uniform early-out.
// ---------------------------------------------------------------------------
__global__ __launch_bounds__(256) void phase3_dominate(
    const float* __restrict__ meanArr, const float* __restrict__ riskArr,
    float* __restrict__ out) {
  __shared__ __align__(16) float sm[N_];
  __shared__ __align__(16) float sr[N_];
  const int b  = blockIdx.x >> 4;   // 16 i-blocks per batch
  const int ib = blockIdx.x & 15;

  const float* mb = meanArr + (size_t)b * N_;
  const float* rb = riskArr + (size_t)b * N_;
  for (int j = threadIdx.x; j < N_; j += 256) {
    sm[j] = mb[j];
    sr[j] = rb[j];
  }
  __syncthreads();

  const int i = ib * 256 + threadIdx.x;
  const float mi = sm[i];
  const float ri = sr[i];

  int dom = 0;
  for (int jb = 0; jb < N_; jb += 256) {
#pragma unroll
    for (int j = jb; j < jb + 256; j += 4) {
      float4 mj = *(const float4*)&sm[j];
      float4 rj = *(const float4*)&sr[j];
      dom |= ((mj.x > mi) & (rj.x < ri));
      dom |= ((mj.y > mi) & (rj.y < ri));
      dom |= ((mj.z > mi) & (rj.z < ri));
      dom |= ((mj.w > mi) & (rj.w < ri));
    }
    if (__all(dom)) break;  // whole wave dominated -> done
  }
  out[(size_t)b * N_ + i] = dom ? 0.0f : 1.0f;
}

// ---------------------------------------------------------------------------
extern "C" void kernel_launch(void* const* d_in, const int* in_sizes, int n_in,
                              void* d_out, int out_size, void* d_ws,
                              size_t ws_size, hipStream_t stream) {
  (void)in_sizes; (void)n_in; (void)out_size; (void)ws_size;

  const float* x = (const float*)d_in[0];
  float* out = (float*)d_out;

  // Workspace layout (floats): 2 * TSPLIT*B*N partials + 2 * B*N stats
  float* sumPart = (float*)d_ws;
  float* sqPart  = sumPart + (size_t)TSPLIT * B_ * N_;
  float* meanArr = sqPart + (size_t)TSPLIT * B_ * N_;
  float* riskArr = meanArr + (size_t)B_ * N_;

  dim3 g1(B_ * (N_ / 256), TSPLIT);  // 128 x 8 = 1024 blocks
  phase1_partials<<<g1, 256, 0, stream>>>(x, sumPart, sqPart);

  phase2_combine<<<(B_ * N_) / 256, 256, 0, stream>>>(sumPart, sqPart,
                                                      meanArr, riskArr);

  phase3_dominate<<<B_ * (N_ / 256), 256, 0, stream>>>(meanArr, riskArr, out);
}